// RetentionConv_42545946034325
// MI455X (gfx1250) — compile-verified
//
#include <hip/hip_runtime.h>
#include <math.h>

// Problem constants (match the reference file)
#define BATCH 2
#define LEN   1024
#define DIMM  512
#define HEADS 8
#define DQKV  32
#define HD    (HEADS * DQKV)     // 256
#define MROWS (BATCH * LEN)      // 2048

typedef float v2f __attribute__((ext_vector_type(2)));
typedef float v8f __attribute__((ext_vector_type(8)));

// ---------------------------------------------------------------------------
// GEMM:  Y[M,N] = A[M,K] @ W[N,K]^T + bias[N]
// One 16x16 output tile per wave, K-steps of 4 via V_WMMA_F32_16X16X4_F32.
// Grid: x = M/16 tiles, y = N/128 groups; block = 256 threads (8 waves).
// Fragment layouts per CDNA5 ISA 7.12.2:
//   A 16x4 : lane L -> M=L&15, K-pair base (L>>4)*2  (2 VGPRs)
//   B 4x16 : lane L -> N=L&15, K rows (L>>4)*2, (L>>4)*2+1
//   C/D    : reg r  -> row r + (L>>4)*8, col L&15
// ---------------------------------------------------------------------------
__global__ __launch_bounds__(256)
void wmma_gemm_bias_f32(const float* __restrict__ A,
                        const float* __restrict__ W,
                        const float* __restrict__ bias,
                        float* __restrict__ Y,
                        int M, int N, int K)
{
    const int lane = threadIdx.x & 31;
    const int wave = threadIdx.x >> 5;
    const int l15  = lane & 15;
    const int half = lane >> 4;          // 0 or 1

    const int m0 = blockIdx.x * 16;
    const int n0 = (blockIdx.y * 8 + wave) * 16;

    const float* __restrict__ arow = A + (size_t)(m0 + l15) * K + half * 2;
    const float* __restrict__ wrow = W + (size_t)(n0 + l15) * K + half * 2;

    v8f acc = {0.f, 0.f, 0.f, 0.f, 0.f, 0.f, 0.f, 0.f};

#pragma unroll 4
    for (int k = 0; k < K; k += 4) {
        v2f afrag = *(const v2f*)(arow + k);
        v2f bfrag = *(const v2f*)(wrow + k);
        // D = A(16x4) * B(4x16) + C, fp32 in / fp32 accumulate
        acc = __builtin_amdgcn_wmma_f32_16x16x4_f32(
            /*neg_a=*/false, afrag,
            /*neg_b=*/false, bfrag,
            /*c_mod=*/(short)0, acc,
            /*reuse_a=*/false, /*reuse_b=*/false);
    }

    const int col  = n0 + l15;
    const float bv = bias[col];
    const int rbase = m0 + half * 8;
#pragma unroll
    for (int r = 0; r < 8; ++r) {
        Y[(size_t)(rbase + r) * N + col] = acc[r] + bv;
    }
}

// ---------------------------------------------------------------------------
// Retention recurrence (replaces the FFT conv exactly):
//   ph = p/|p| * exp(-|p|)
//   S[t]   = ph * S[t-1] + k[t,i]*v[t,j]          (complex, S[-1]=0)
//   R[t]   = Re( phazor_init * S[t] + last_conv_init * ph^(t+1) )
//   out_h[t,j] = sum_i q[t,i] * R[t,i,j]
// One block per (b,h); wave w <-> j, lane <-> i; sum over i is a wave32
// shuffle-xor butterfly (off the serial S critical path).
// ---------------------------------------------------------------------------
__global__ __launch_bounds__(1024)
void retention_recurrence(const float* __restrict__ q,
                          const float* __restrict__ k,
                          const float* __restrict__ v,
                          const float* __restrict__ phazor,       // complex64 [H,32,32]
                          const float* __restrict__ phazor_init,  // complex64 [H,32,32]
                          const float* __restrict__ last_conv,    // complex64 [H,32,32]
                          float* __restrict__ out_h)              // [B,L,H,32]
{
    const int bh = blockIdx.x;        // 0 .. B*H-1
    const int bb = bh / HEADS;
    const int h  = bh % HEADS;
    const int j  = threadIdx.x >> 5;  // wave id -> output column
    const int i  = threadIdx.x & 31;  // lane -> row

    const int idx = (h * DQKV + i) * DQKV + j;   // per-(h,i,j) element

    // decayed phazor on the unit ray: p/|p| * exp(-|p|)
    const float pr  = phazor[2 * idx + 0];
    const float pim = phazor[2 * idx + 1];
    const float mag = sqrtf(pr * pr + pim * pim);
    const float scl = expf(-mag) / mag;
    const float ar  = pr * scl;       // Re(ph)
    const float ai  = pim * scl;      // Im(ph)

    const float pir = phazor_init[2 * idx + 0];
    const float pii = phazor_init[2 * idx + 1];
    const float lcr = last_conv[2 * idx + 0];
    const float lci = last_conv[2 * idx + 1];

    float Sr = 0.f, Si = 0.f;         // complex recurrent state
    float Pr = ar, Pi = ai;           // ph^(t+1), starts at ph for t=0

    const size_t rowstride = HD;      // floats per (b,t)
    const float* __restrict__ kb = k + ((size_t)bb * LEN) * rowstride + h * DQKV;
    const float* __restrict__ vb = v + ((size_t)bb * LEN) * rowstride + h * DQKV;
    const float* __restrict__ qb = q + ((size_t)bb * LEN) * rowstride + h * DQKV;
    float* __restrict__ ob = out_h + ((size_t)bb * LEN) * rowstride + h * DQKV;

    for (int t = 0; t < LEN; ++t) {
        const size_t off = (size_t)t * rowstride;
        const float kv = kb[off + i] * vb[off + j];

        // S = ph*S + kv   (complex FMA chain — the serial critical path)
        const float nSr = fmaf(ar, Sr, fmaf(-ai, Si, kv));
        const float nSi = fmaf(ar, Si, ai * Sr);
        Sr = nSr; Si = nSi;

        // carry of initial state: Re(last_conv_init * ph^(t+1))
        const float carry = fmaf(lcr, Pr, -lci * Pi);
        // advance P = ph^(t+2)
        const float nPr = fmaf(ar, Pr, -ai * Pi);
        const float nPi = fmaf(ar, Pi, ai * Pr);
        Pr = nPr; Pi = nPi;

        // Re(phazor_init * S) + carry
        const float R = fmaf(pir, Sr, fmaf(-pii, Si, carry));

        // contract with q over i (lanes) — wave32 butterfly reduction
        float contrib = qb[off + i] * R;
#pragma unroll
        for (int m = 16; m > 0; m >>= 1)
            contrib += __shfl_xor(contrib, m, 32);

        if (i == 0)
            ob[off + j] = contrib;
    }
}

// ---------------------------------------------------------------------------
// Launch: 3 input projections -> recurrence -> output projection.
// Workspace: q | k | v | out_h  (each 2048*256 f32 = 2 MB, total 8 MB).
// ---------------------------------------------------------------------------
extern "C" void kernel_launch(void* const* d_in, const int* in_sizes, int n_in,
                              void* d_out, int out_size, void* d_ws, size_t ws_size,
                              hipStream_t stream)
{
    (void)in_sizes; (void)n_in; (void)out_size; (void)ws_size;

    const float* x      = (const float*)d_in[0];
    const float* wq_w   = (const float*)d_in[1];
    const float* wq_b   = (const float*)d_in[2];
    const float* wk_w   = (const float*)d_in[3];
    const float* wk_b   = (const float*)d_in[4];
    const float* wv_w   = (const float*)d_in[5];
    const float* wv_b   = (const float*)d_in[6];
    const float* wout_w = (const float*)d_in[7];
    const float* wout_b = (const float*)d_in[8];
    const float* phz    = (const float*)d_in[9];   // complex64 -> float pairs
    const float* phz0   = (const float*)d_in[10];
    const float* lconv  = (const float*)d_in[11];

    float* q  = (float*)d_ws;
    float* k  = q + (size_t)MROWS * HD;
    float* v  = k + (size_t)MROWS * HD;
    float* oh = v + (size_t)MROWS * HD;

    const dim3 blk(256);
    // q/k/v projections: M=2048, N=256, K=512 -> grid (128, 2)
    const dim3 gproj(MROWS / 16, HD / 128);
    wmma_gemm_bias_f32<<<gproj, blk, 0, stream>>>(x, wq_w, wq_b, q, MROWS, HD, DIMM);
    wmma_gemm_bias_f32<<<gproj, blk, 0, stream>>>(x, wk_w, wk_b, k, MROWS, HD, DIMM);
    wmma_gemm_bias_f32<<<gproj, blk, 0, stream>>>(x, wv_w, wv_b, v, MROWS, HD, DIMM);

    // recurrence: one block per (b,h), 32 waves of 32 lanes
    retention_recurrence<<<dim3(BATCH * HEADS), dim3(1024), 0, stream>>>(
        q, k, v, phz, phz0, lconv, oh);

    // output projection: M=2048, N=512, K=256 -> grid (128, 4)
    const dim3 gout(MROWS / 16, DIMM / 128);
    wmma_gemm_bias_f32<<<gout, blk, 0, stream>>>(oh, wout_w, wout_b, (float*)d_out,
                                                 MROWS, DIMM, HD);
}